// CTCLossManual_50113678410612
// MI455X (gfx1250) — compile-verified
//
#include <hip/hip_runtime.h>
#include <math.h>

#define N_CLS 8192
#define NEGV (-1.0e30f)
#define MAX_T 4096
#define MAX_S 1056  // >= 2*512+1, padded

// ---- CDNA5 async global->LDS staging (guarded; falls back to vector copy) ----
#if defined(__gfx1250__) && __has_builtin(__builtin_amdgcn_global_load_async_to_lds_b128)
#define HAVE_ASYNC_LDS 1
#else
#define HAVE_ASYNC_LDS 0
#endif

typedef int v4i __attribute__((ext_vector_type(4)));
#if defined(__gfx1250__)
typedef __attribute__((address_space(1))) v4i* g_v4i_p;  // global
typedef __attribute__((address_space(3))) v4i* l_v4i_p;  // LDS
#endif

__device__ __forceinline__ void wait_async0() {
#if defined(__gfx1250__)
#if __has_builtin(__builtin_amdgcn_s_wait_asynccnt)
    __builtin_amdgcn_s_wait_asynccnt(0);
#else
    asm volatile("s_wait_asynccnt 0" ::: "memory");
#endif
#endif
}

// ---------------------------------------------------------------------------
// Kernel A: one block per time step t. Stage the 32KB row into LDS via 8
// straight-line async b128 copies per thread (CDNA5 TDM-lite path), then
// compute lse[t] = max + log(sum(exp(x - max))).
// Bandwidth-bound: 128 MB total -> ~5.5us floor at 23.3 TB/s.
// ---------------------------------------------------------------------------
__global__ __launch_bounds__(256) void ctc_row_lse(const float* __restrict__ logits,
                                                   float* __restrict__ lse) {
    __shared__ float row[N_CLS];
    __shared__ float redm[8];
    __shared__ float reds[8];
    const int tid = threadIdx.x;  // blockDim.x == 256 exactly
    const size_t base = (size_t)blockIdx.x * N_CLS;
    const float4* g4 = (const float4*)(logits + base);
    float4* r4 = (float4*)row;

#if HAVE_ASYNC_LDS
#pragma unroll
    for (int k = 0; k < 8; ++k) {
        const int i = tid + k * 256;  // 8 * 256 = 2048 = N_CLS/4, no guard needed
        __builtin_amdgcn_global_load_async_to_lds_b128(
            (g_v4i_p)(g4 + i), (l_v4i_p)(r4 + i), /*offset=*/0, /*cpol=*/0);
    }
    wait_async0();
#else
#pragma unroll
    for (int k = 0; k < 8; ++k) {
        const int i = tid + k * 256;
        r4[i] = g4[i];
    }
#endif
    __syncthreads();

    // max over row
    float m = NEGV;
#pragma unroll 8
    for (int k = 0; k < N_CLS / 256; ++k) m = fmaxf(m, row[tid + k * 256]);
    for (int off = 16; off >= 1; off >>= 1) m = fmaxf(m, __shfl_xor(m, off, 32));
    if ((tid & 31) == 0) redm[tid >> 5] = m;
    __syncthreads();
    const float bm = fmaxf(fmaxf(fmaxf(redm[0], redm[1]), fmaxf(redm[2], redm[3])),
                           fmaxf(fmaxf(redm[4], redm[5]), fmaxf(redm[6], redm[7])));

    // sum(exp(x - max))
    float s = 0.0f;
#pragma unroll 8
    for (int k = 0; k < N_CLS / 256; ++k) s += __expf(row[tid + k * 256] - bm);
    for (int off = 16; off >= 1; off >>= 1) s += __shfl_xor(s, off, 32);
    if ((tid & 31) == 0) reds[tid >> 5] = s;
    __syncthreads();
    if (tid == 0) {
        float tot = ((reds[0] + reds[1]) + (reds[2] + reds[3])) +
                    ((reds[4] + reds[5]) + (reds[6] + reds[7]));
        lse[blockIdx.x] = bm + __logf(tot);
    }
}

// ---------------------------------------------------------------------------
// Kernel B: single 1024-thread workgroup runs the sequential alpha recursion.
// Each thread owns state(s) s = tid (+1024). alpha double-buffered in LDS
// (one barrier per step). lse[] cached fully in LDS. Gather of
// logits[t, tw[s]] software-pipelined one step ahead (hits L2; the whole
// 128 MB logits tensor fits in the 192 MB L2 after kernel A streams it).
// ---------------------------------------------------------------------------
__global__ __launch_bounds__(1024) void ctc_alpha(const float* __restrict__ logits,
                                                  const float* __restrict__ lse,
                                                  const int* __restrict__ targets,
                                                  float* __restrict__ out,
                                                  int T, int L) {
    __shared__ float lse_s[MAX_T];
    __shared__ float abuf[2][MAX_S + 2];  // [0..1] = NEG pads, state s at [s+2]
    const int tid = threadIdx.x;
    const int S = 2 * L + 1;

    for (int i = tid; i < T && i < MAX_T; i += 1024) lse_s[i] = lse[i];
    for (int i = tid; i < MAX_S + 2; i += 1024) {
        abuf[0][i] = NEGV;
        abuf[1][i] = NEGV;
    }

    // per-thread state bookkeeping (at most 2 states per thread)
    int nmine = 0;
    int ss[2];
    int tw[2];
    bool allow[2];
    for (int s = tid; s < S && nmine < 2; s += 1024) {
        int c = (s & 1) ? targets[s >> 1] : 0;
        ss[nmine] = s;
        tw[nmine] = c;
        bool a = ((s & 1) != 0) && (c != 0);
        if (s >= 3) a = a && (c != targets[(s >> 1) - 1]);
        allow[nmine] = a;
        nmine++;
    }
    __syncthreads();

    // alpha0
    for (int k = 0; k < nmine; ++k) {
        if (ss[k] == 0) abuf[0][2] = logits[0] - lse_s[0];
        if (ss[k] == 1) abuf[0][3] = logits[tw[k]] - lse_s[0];
    }
    __syncthreads();

    // prefetch gather for t=1
    float xn[2] = {0.0f, 0.0f};
    if (T > 1) {
        for (int k = 0; k < nmine; ++k) xn[k] = logits[(size_t)N_CLS + tw[k]];
    }

    for (int t = 1; t < T; ++t) {
        const float* __restrict__ prev = abuf[(t - 1) & 1];
        float* __restrict__ cur = abuf[t & 1];
        const float lt = lse_s[t];
        const size_t nbase = (size_t)(t + 1) * N_CLS;
        float res[2];
        for (int k = 0; k < nmine; ++k) {
            float x = xn[k];
            if (t + 1 < T) xn[k] = logits[nbase + tw[k]];  // pipeline next gather
            const int s = ss[k];
            float a0 = prev[s + 2];
            float a1 = prev[s + 1];
            float a2 = allow[k] ? prev[s] : NEGV;
            float m = fmaxf(a0, fmaxf(a1, a2));
            float r = m + __logf(__expf(a0 - m) + __expf(a1 - m) + __expf(a2 - m));
            res[k] = r + (x - lt);
        }
        for (int k = 0; k < nmine; ++k) cur[ss[k] + 2] = res[k];
        __syncthreads();
    }

    if (tid == 0) {
        const float* fin = abuf[(T - 1) & 1];
        float a = fin[2 + S - 2];
        float b = fin[2 + S - 1];
        float m = fmaxf(a, b);
        out[0] = -(m + __logf(__expf(a - m) + __expf(b - m)));
    }
}

// ---------------------------------------------------------------------------
extern "C" void kernel_launch(void* const* d_in, const int* in_sizes, int n_in,
                              void* d_out, int out_size, void* d_ws, size_t ws_size,
                              hipStream_t stream) {
    const float* logits = (const float*)d_in[0];
    const int* targets = (const int*)d_in[1];
    const int T = in_sizes[0] / N_CLS;  // 4096
    const int L = in_sizes[1];          // 512
    float* lse = (float*)d_ws;          // T floats of scratch

    ctc_row_lse<<<T, 256, 0, stream>>>(logits, lse);
    ctc_alpha<<<1, 1024, 0, stream>>>(logits, lse, targets, (float*)d_out, T, L);
}